// LorentzNet_17257178595648
// MI455X (gfx1250) — compile-verified
//
#include <hip/hip_runtime.h>

// ---------------- model constants ----------------
constexpr int NHID   = 72;
constexpr int NHP    = 80;                 // hidden padded to 5x16
constexpr int NSC    = 8;
constexpr int NBATCH = 32;
constexpr int NNODE  = 128;
constexpr int NEDGE  = NNODE * (NNODE - 1);   // 16256
constexpr int BE     = NBATCH * NEDGE;        // 520192 edge rows
constexpr int NROWS  = NBATCH * NNODE;        // 4096 node rows
constexpr int EK     = 2 * NHID + 2;          // 146 edge-feat dim
constexpr int EKP    = 148;                   // padded to multiple of 4
constexpr int NK     = 2 * NHID + NSC;        // 152 node-feat dim
constexpr int LDE    = 149;                   // edge tile stride (>=148, odd)
constexpr int LDN    = 153;                   // node-pass1 tile stride (>=152, odd)
constexpr int LDP    = 81;                    // node-pass2 tile stride (>=80, odd)
constexpr int NLAYERS = 6;
constexpr float BN_EPS   = 1e-5f;
constexpr float C_WEIGHT = 0.001f;

typedef __attribute__((ext_vector_type(2))) float v2f;
typedef __attribute__((ext_vector_type(8))) float v8f;

// ---------------- CDNA5 fp32 WMMA (16x16xK=4), D = A*B + C ----------------
__device__ __forceinline__ v8f wmma4(v2f a, v2f b, v8f c) {
  return __builtin_amdgcn_wmma_f32_16x16x4_f32(false, a, false, b, (short)0, c, false, false);
}

// A-frag (32-bit A 16x4): lanes 0-15 rows m, K-half 0; lanes 16-31 rows m, K-half 1.
// C/D: VGPR r, lanes 0-15 -> row r; lanes 16-31 -> row r+8; col = lane&15.
__device__ __forceinline__ void gemm_tile(const float* __restrict__ tile, int ld,
                                          const float* __restrict__ Wp,
                                          int Kp, v8f* acc, int lane) {
  const int m  = lane & 15;
  const int kh = (lane >> 4) << 1;
  const int n  = lane & 15;
  for (int k = 0; k < Kp; k += 4) {
    v2f a;
    a.x = tile[m * ld + k + kh];
    a.y = tile[m * ld + k + kh + 1];
#pragma unroll
    for (int t = 0; t < 5; ++t) {
      const float* wp = Wp + (k + kh) * NHP + 16 * t + n;
      v2f b;
      b.x = wp[0];
      b.y = wp[NHP];
      acc[t] = wmma4(a, b, acc[t]);
    }
  }
}

// ---------------- CDNA5 async global->LDS staging (ASYNCcnt) ----------------
#if __has_builtin(__builtin_amdgcn_global_load_async_to_lds_b128) && \
    __has_builtin(__builtin_amdgcn_s_wait_asynccnt)
#define USE_ASYNC_LDS 1
typedef int v4i_vs __attribute__((__vector_size__(16)));
typedef __attribute__((address_space(1))) v4i_vs* as1_v4i_ptr;
typedef __attribute__((address_space(3))) v4i_vs* as3_v4i_ptr;
#endif

__device__ __forceinline__ void stage_to_lds(const float* __restrict__ src,
                                             float* dst_lds, int nFloats,
                                             int tid, int nthr) {
#ifdef USE_ASYNC_LDS
  for (int o = tid * 4; o < nFloats; o += nthr * 4) {
    __builtin_amdgcn_global_load_async_to_lds_b128(
        (as1_v4i_ptr)(src + o), (as3_v4i_ptr)(dst_lds + o), 0, 0);
  }
#else
  for (int o = tid * 4; o < nFloats; o += nthr * 4) {
    float4 v = *(const float4*)(src + o);
    *(float4*)(dst_lds + o) = v;
  }
#endif
}
__device__ __forceinline__ void stage_wait() {
#ifdef USE_ASYNC_LDS
  __builtin_amdgcn_s_wait_asynccnt(0);
#endif
}

// ---------------- helpers ----------------
__device__ __forceinline__ void edge_decode(int R, int& b, int& i, int& j) {
  b = R / NEDGE;
  int e = R - b * NEDGE;
  i = e / (NNODE - 1);
  int jj = e - i * (NNODE - 1);
  j = jj + (jj >= i);
}
__device__ __forceinline__ float psi_f(float z) {
  return copysignf(log1pf(fabsf(z)), z);
}

__device__ void fill_feat(float* tile, int tileBase, const float* __restrict__ h,
                          const float* __restrict__ x, int lane) {
  for (int idx = lane; idx < 16 * EKP; idx += 32) {
    int m = idx / EKP, k = idx - m * EKP;
    int b, i, j;
    edge_decode(tileBase + m, b, i, j);
    float v;
    if (k < NHID) v = h[(b * NNODE + i) * NHID + k];
    else if (k < 2 * NHID) v = h[(b * NNODE + j) * NHID + (k - NHID)];
    else if (k < EK) {
      const float* xi = x + (b * NNODE + i) * 4;
      const float* xj = x + (b * NNODE + j) * 4;
      if (k == 2 * NHID) {
        float d0 = xi[0] - xj[0], d1 = xi[1] - xj[1], d2 = xi[2] - xj[2], d3 = xi[3] - xj[3];
        v = psi_f(d0 * d0 - d1 * d1 - d2 * d2 - d3 * d3);
      } else {
        v = psi_f(xi[0] * xj[0] - xi[1] * xj[1] - xi[2] * xj[2] - xi[3] * xj[3]);
      }
    } else v = 0.f;
    tile[m * LDE + k] = v;
  }
}

// ---------------- utility kernels ----------------
__global__ __launch_bounds__(256) void zero_kernel(float* p, int n) {
  int i = blockIdx.x * blockDim.x + threadIdx.x;
  if (i < n) p[i] = 0.f;
}

__global__ __launch_bounds__(256) void pack_kernel(const float* __restrict__ src,
                                                   float* __restrict__ dst, int K, int Kp) {
  int idx = blockIdx.x * blockDim.x + threadIdx.x;
  if (idx < Kp * NHP) {
    int k = idx / NHP, n = idx - k * NHP;
    dst[idx] = (k < K && n < NHID) ? src[k * NHID + n] : 0.f;
  }
}

__global__ __launch_bounds__(256) void embed_kernel(const float* __restrict__ scal,
                                                    const float* __restrict__ x_in,
                                                    const float* __restrict__ eW,
                                                    const float* __restrict__ eb,
                                                    float* __restrict__ h, float* __restrict__ x) {
  int idx = blockIdx.x * blockDim.x + threadIdx.x;
  if (idx < NROWS * NHID) {
    int row = idx / NHID, c = idx - row * NHID;
    float s = eb[c];
#pragma unroll
    for (int k = 0; k < NSC; ++k) s += scal[row * NSC + k] * eW[k * NHID + c];
    h[idx] = s;
  }
  if (idx < NROWS * 4) x[idx] = x_in[idx];
}

__global__ __launch_bounds__(128) void bn_finalize(const float* __restrict__ sums,
                                                   const float* __restrict__ g,
                                                   const float* __restrict__ bb,
                                                   float* __restrict__ ac, float inv_n) {
  int c = threadIdx.x;
  if (c >= NHP) return;
  float a = 0.f, cc = 0.f;
  if (c < NHID) {
    float mu  = sums[c] * inv_n;
    float var = sums[NHP + c] * inv_n - mu * mu;
    a  = g[c] * rsqrtf(var + BN_EPS);
    cc = bb[c] - mu * a;
  }
  ac[c] = a;
  ac[NHP + c] = cc;
}

// ---------------- edge pass 1: z = feat@e1_W ; accumulate BN sums ----------------
__global__ __launch_bounds__(128) void edge_pass1(const float* __restrict__ h,
                                                  const float* __restrict__ x,
                                                  const float* __restrict__ e1p,
                                                  float* __restrict__ sums) {
  __shared__ float tile[4][16 * LDE];
  __shared__ float s1s[NHP], s2s[NHP];
  const int w = threadIdx.x >> 5, lane = threadIdx.x & 31;
  if (threadIdx.x < NHP) { s1s[threadIdx.x] = 0.f; s2s[threadIdx.x] = 0.f; }
  for (int off = threadIdx.x * 16; off < EKP * NHP; off += 128 * 16)
    __builtin_prefetch(&e1p[off], 0, 3);
  const int tileBase = (blockIdx.x * 4 + w) * 16;
  fill_feat(tile[w], tileBase, h, x, lane);
  __syncthreads();
  v8f acc[5] = {};
  gemm_tile(tile[w], LDE, e1p, EKP, acc, lane);
#pragma unroll
  for (int t = 0; t < 5; ++t) {
    float s1 = 0.f, s2 = 0.f;
#pragma unroll
    for (int r = 0; r < 8; ++r) { float z = acc[t][r]; s1 += z; s2 += z * z; }
    s1 += __shfl_xor(s1, 16);
    s2 += __shfl_xor(s2, 16);
    if (lane < 16) {
      atomicAdd(&s1s[16 * t + lane], s1);
      atomicAdd(&s2s[16 * t + lane], s2);
    }
  }
  __syncthreads();
  if (threadIdx.x < NHP) {
    atomicAdd(&sums[threadIdx.x], s1s[threadIdx.x]);
    atomicAdd(&sums[NHP + threadIdx.x], s2s[threadIdx.x]);
  }
}

// ---------------- edge pass 2: fused edge MLP, LDS-staged square weights -----
__global__ __launch_bounds__(128) void edge_pass2(
    const float* __restrict__ h, const float* __restrict__ x,
    const float* __restrict__ e1p, const float* __restrict__ ac,
    const float* __restrict__ e2p, const float* __restrict__ e2b,
    const float* __restrict__ mW, const float* __restrict__ mb,
    const float* __restrict__ x1p, const float* __restrict__ x1b,
    const float* __restrict__ x2W,
    float* __restrict__ agg_h, float* __restrict__ agg_x, int lastLayer) {
  __shared__ float tile[4][16 * LDE];
  __shared__ __align__(16) float wsq[NHP * NHP];   // staged e2 / x1 weights
  const int w = threadIdx.x >> 5, lane = threadIdx.x & 31;
  const int n = lane & 15, g = lane >> 4;
  const int tileBase = (blockIdx.x * 4 + w) * 16;
  // kick off async DMA of e2 weights into LDS; overlap with feat build + GEMM1
  stage_to_lds(e2p, wsq, NHP * NHP, threadIdx.x, 128);
  fill_feat(tile[w], tileBase, h, x, lane);
  __syncthreads();
  v8f acc[5] = {};
  gemm_tile(tile[w], LDE, e1p, EKP, acc, lane);
  __syncthreads();
  // z = relu(a*z + c); stage transposed tile (padded cols -> 0)
#pragma unroll
  for (int t = 0; t < 5; ++t) {
    int c = 16 * t + n;
    float a_ = ac[c], c_ = ac[NHP + c];
#pragma unroll
    for (int r = 0; r < 8; ++r)
      tile[w][(r + 8 * g) * LDE + c] = fmaxf(acc[t][r] * a_ + c_, 0.f);
  }
  stage_wait();          // each wave's async e2 loads landed before it signals
  __syncthreads();
  // m = relu(z @ e2 + b)   (B-fragments from LDS)
  v8f macc[5] = {};
  gemm_tile(tile[w], LDE, wsq, NHP, macc, lane);
#pragma unroll
  for (int t = 0; t < 5; ++t) {
    int c = 16 * t + n;
    float bias = (c < NHID) ? e2b[c] : 0.f;
#pragma unroll
    for (int r = 0; r < 8; ++r) macc[t][r] = fmaxf(macc[t][r] + bias, 0.f);
  }
  if (!lastLayer) {
    __syncthreads();     // everyone done reading wsq(e2) -> safe to restage
    stage_to_lds(x1p, wsq, NHP * NHP, threadIdx.x, 128);
  }
  // w = sigmoid(m @ mW + mb); m *= w   (16-lane butterfly, overlaps x1 DMA)
  {
    float mwv[5];
#pragma unroll
    for (int t = 0; t < 5; ++t) { int c = 16 * t + n; mwv[t] = (c < NHID) ? mW[c] : 0.f; }
    float mbv = mb[0];
#pragma unroll
    for (int r = 0; r < 8; ++r) {
      float p = 0.f;
#pragma unroll
      for (int t = 0; t < 5; ++t) p += macc[t][r] * mwv[t];
      p += __shfl_xor(p, 1); p += __shfl_xor(p, 2);
      p += __shfl_xor(p, 4); p += __shfl_xor(p, 8);
      float wv = 1.f / (1.f + __expf(-(p + mbv)));
#pragma unroll
      for (int t = 0; t < 5; ++t) macc[t][r] *= wv;
    }
  }
  // scatter m -> agg_h[b, i, :]
#pragma unroll
  for (int r = 0; r < 8; ++r) {
    int b, i, j;
    edge_decode(tileBase + r + 8 * g, b, i, j);
#pragma unroll
    for (int t = 0; t < 5; ++t) {
      int c = 16 * t + n;
      if (c < NHID) atomicAdd(&agg_h[(b * NNODE + i) * NHID + c], macc[t][r]);
    }
  }
  if (!lastLayer) {
#pragma unroll
    for (int t = 0; t < 5; ++t) {
      int c = 16 * t + n;
#pragma unroll
      for (int r = 0; r < 8; ++r) tile[w][(r + 8 * g) * LDE + c] = macc[t][r];
    }
    stage_wait();
    __syncthreads();
    v8f tacc[5] = {};
    gemm_tile(tile[w], LDE, wsq, NHP, tacc, lane);   // B = x1 from LDS
    float x2v[5];
#pragma unroll
    for (int t = 0; t < 5; ++t) { int c = 16 * t + n; x2v[t] = (c < NHID) ? x2W[c] : 0.f; }
#pragma unroll
    for (int r = 0; r < 8; ++r) {
      float p = 0.f;
#pragma unroll
      for (int t = 0; t < 5; ++t) {
        int c = 16 * t + n;
        float bias = (c < NHID) ? x1b[c] : 0.f;
        p += fmaxf(tacc[t][r] + bias, 0.f) * x2v[t];
      }
      p += __shfl_xor(p, 1); p += __shfl_xor(p, 2);
      p += __shfl_xor(p, 4); p += __shfl_xor(p, 8);
      if (n < 4) {
        int b, i, j;
        edge_decode(tileBase + r + 8 * g, b, i, j);
        float d = x[(b * NNODE + i) * 4 + n] - x[(b * NNODE + j) * 4 + n];
        float tr = fminf(fmaxf(d * p, -100.f), 100.f);
        atomicAdd(&agg_x[(b * NNODE + i) * 4 + n], tr);
      }
    }
  }
}

__global__ __launch_bounds__(256) void x_update(float* __restrict__ x,
                                                const float* __restrict__ agg_x) {
  int i = blockIdx.x * blockDim.x + threadIdx.x;
  if (i < NROWS * 4) x[i] += agg_x[i] * (C_WEIGHT / 127.0f);
}

// ---------------- node pass 1: u = hf@h1 + b, store u, BN sums ----------------
__global__ __launch_bounds__(128) void node_pass1(const float* __restrict__ h,
                                                  const float* __restrict__ agg,
                                                  const float* __restrict__ scal,
                                                  const float* __restrict__ h1p,
                                                  const float* __restrict__ h1b,
                                                  float* __restrict__ u, float* __restrict__ sums) {
  __shared__ float tile[4][16 * LDN];
  __shared__ float s1s[NHP], s2s[NHP];
  const int w = threadIdx.x >> 5, lane = threadIdx.x & 31;
  if (threadIdx.x < NHP) { s1s[threadIdx.x] = 0.f; s2s[threadIdx.x] = 0.f; }
  const int tileBase = (blockIdx.x * 4 + w) * 16;
  for (int idx = lane; idx < 16 * NK; idx += 32) {
    int m = idx / NK, k = idx - m * NK;
    int row = tileBase + m;
    float v;
    if (k < NHID) v = h[row * NHID + k];
    else if (k < 2 * NHID) v = agg[row * NHID + (k - NHID)];
    else v = scal[row * NSC + (k - 2 * NHID)];
    tile[w][m * LDN + k] = v;
  }
  __syncthreads();
  v8f acc[5] = {};
  gemm_tile(tile[w], LDN, h1p, NK, acc, lane);
  const int n = lane & 15, g = lane >> 4;
#pragma unroll
  for (int t = 0; t < 5; ++t) {
    int c = 16 * t + n;
    float bias = (c < NHID) ? h1b[c] : 0.f;
    float s1 = 0.f, s2 = 0.f;
#pragma unroll
    for (int r = 0; r < 8; ++r) {
      float uv = acc[t][r] + bias;
      u[(tileBase + r + 8 * g) * NHP + c] = uv;
      s1 += uv; s2 += uv * uv;
    }
    s1 += __shfl_xor(s1, 16);
    s2 += __shfl_xor(s2, 16);
    if (lane < 16) { atomicAdd(&s1s[c], s1); atomicAdd(&s2s[c], s2); }
  }
  __syncthreads();
  if (threadIdx.x < NHP) {
    atomicAdd(&sums[threadIdx.x], s1s[threadIdx.x]);
    atomicAdd(&sums[NHP + threadIdx.x], s2s[threadIdx.x]);
  }
}

// ---------------- node pass 2: h += relu(bn(u)) @ h2 + b ----------------
__global__ __launch_bounds__(128) void node_pass2(const float* __restrict__ u,
                                                  const float* __restrict__ ac,
                                                  const float* __restrict__ h2p,
                                                  const float* __restrict__ h2b,
                                                  float* __restrict__ h) {
  __shared__ float tile[4][16 * LDP];
  __shared__ __align__(16) float wsq[NHP * NHP];
  const int w = threadIdx.x >> 5, lane = threadIdx.x & 31;
  const int tileBase = (blockIdx.x * 4 + w) * 16;
  stage_to_lds(h2p, wsq, NHP * NHP, threadIdx.x, 128);
  for (int idx = lane; idx < 16 * NHP; idx += 32) {
    int m = idx / NHP, k = idx - m * NHP;
    float uv = u[(tileBase + m) * NHP + k];
    tile[w][m * LDP + k] = fmaxf(uv * ac[k] + ac[NHP + k], 0.f);
  }
  stage_wait();
  __syncthreads();
  v8f acc[5] = {};
  gemm_tile(tile[w], LDP, wsq, NHP, acc, lane);
  const int n = lane & 15, g = lane >> 4;
#pragma unroll
  for (int t = 0; t < 5; ++t) {
    int c = 16 * t + n;
    if (c < NHID) {
      float bias = h2b[c];
#pragma unroll
      for (int r = 0; r < 8; ++r) {
        int row = tileBase + r + 8 * g;
        h[row * NHID + c] += acc[t][r] + bias;
      }
    }
  }
}

// ---------------- decode (tiny) ----------------
__global__ __launch_bounds__(256) void decode_kernel(const float* __restrict__ h,
                                                     const float* __restrict__ d1W,
                                                     const float* __restrict__ d1b,
                                                     const float* __restrict__ d2W,
                                                     const float* __restrict__ d2b,
                                                     float* __restrict__ out) {
  __shared__ float hg[NBATCH * NHID];
  __shared__ float zz[NBATCH * NHID];
  for (int idx = threadIdx.x; idx < NBATCH * NHID; idx += 256) {
    int b = idx / NHID, c = idx - b * NHID;
    float s = 0.f;
    for (int nn = 0; nn < NNODE; ++nn) s += h[(b * NNODE + nn) * NHID + c];
    hg[idx] = s * (1.0f / NNODE);
  }
  __syncthreads();
  for (int idx = threadIdx.x; idx < NBATCH * NHID; idx += 256) {
    int b = idx / NHID, c = idx - b * NHID;
    float s = d1b[c];
    for (int k = 0; k < NHID; ++k) s += hg[b * NHID + k] * d1W[k * NHID + c];
    zz[idx] = fmaxf(s, 0.f);
  }
  __syncthreads();
  for (int idx = threadIdx.x; idx < NBATCH * 2; idx += 256) {
    int b = idx >> 1, c = idx & 1;
    float s = d2b[c];
    for (int k = 0; k < NHID; ++k) s += zz[b * NHID + k] * d2W[k * 2 + c];
    out[idx] = s;
  }
}

// ---------------- host ----------------
struct LayerP {
  const float *e1W, *ebng, *ebnb, *e2W, *e2b, *mW, *mb, *x1W, *x1b, *x2W;
  const float *h1W, *h1b, *hbng, *hbnb, *h2W, *h2b;
};

extern "C" void kernel_launch(void* const* d_in, const int* in_sizes, int n_in,
                              void* d_out, int out_size, void* d_ws, size_t ws_size,
                              hipStream_t stream) {
  auto F = [&](int idx) -> const float* { return (const float*)d_in[idx]; };
  const float* scal = F(0);   // (32,128,8)
  const float* x_in = F(1);   // (32,128,4)
  const float *embW, *embb, *d1W, *d1b, *d2W, *d2b;
  LayerP L[NLAYERS];
  int p = 2;
  if (n_in > 2 && in_sizes[2] == 576) {   // dict-insertion order: emb, dec1, dec2, layers
    embW = F(p++); embb = F(p++); d1W = F(p++); d1b = F(p++); d2W = F(p++); d2b = F(p++);
    for (int l = 0; l < NLAYERS; ++l) {
      L[l].e1W = F(p++); L[l].ebng = F(p++); L[l].ebnb = F(p++);
      L[l].e2W = F(p++); L[l].e2b = F(p++);
      L[l].mW = F(p++);  L[l].mb = F(p++);
      L[l].x1W = F(p++); L[l].x1b = F(p++); L[l].x2W = F(p++);
      L[l].h1W = F(p++); L[l].h1b = F(p++);
      L[l].hbng = F(p++); L[l].hbnb = F(p++);
      L[l].h2W = F(p++); L[l].h2b = F(p++);
    }
  } else {                                 // jax pytree sorted-key order
    d1W = F(p++); d1b = F(p++); d2W = F(p++); d2b = F(p++); embW = F(p++); embb = F(p++);
    for (int l = 0; l < NLAYERS; ++l) {
      L[l].e1W = F(p++); L[l].e2W = F(p++); L[l].e2b = F(p++);
      L[l].ebnb = F(p++); L[l].ebng = F(p++);
      L[l].h1W = F(p++); L[l].h1b = F(p++); L[l].h2W = F(p++); L[l].h2b = F(p++);
      L[l].hbnb = F(p++); L[l].hbng = F(p++);
      L[l].mW = F(p++); L[l].mb = F(p++);
      L[l].x1W = F(p++); L[l].x1b = F(p++); L[l].x2W = F(p++);
    }
  }

  // workspace layout (floats)
  float* ws   = (float*)d_ws;
  float* h    = ws;                       // 4096*72
  float* xws  = h + NROWS * NHID;         // 4096*4
  float* aggh = xws + NROWS * 4;          // 4096*72
  float* aggx = aggh + NROWS * NHID;      // 4096*4
  float* sums = aggx + NROWS * 4;         // 160
  float* ac   = sums + 2 * NHP;           // 160
  float* u    = ac + 2 * NHP;             // 4096*80
  float* wpk  = u + NROWS * NHP;          // packed weights (16B-aligned offsets)
  const int SZ_E1 = EKP * NHP, SZ_SQ = NHP * NHP, SZ_H1 = NK * NHP;
  const int LPK = SZ_E1 + 3 * SZ_SQ + SZ_H1;   // 43200 floats / layer

  auto grid = [](int n) { return (n + 255) / 256; };

  // pack padded weights
  for (int l = 0; l < NLAYERS; ++l) {
    float* b = wpk + l * LPK;
    pack_kernel<<<grid(SZ_E1), 256, 0, stream>>>(L[l].e1W, b, EK, EKP);
    pack_kernel<<<grid(SZ_SQ), 256, 0, stream>>>(L[l].e2W, b + SZ_E1, NHID, NHP);
    pack_kernel<<<grid(SZ_SQ), 256, 0, stream>>>(L[l].x1W, b + SZ_E1 + SZ_SQ, NHID, NHP);
    pack_kernel<<<grid(SZ_H1), 256, 0, stream>>>(L[l].h1W, b + SZ_E1 + 2 * SZ_SQ, NK, NK);
    pack_kernel<<<grid(SZ_SQ), 256, 0, stream>>>(L[l].h2W, b + SZ_E1 + 2 * SZ_SQ + SZ_H1, NHID, NHP);
  }

  embed_kernel<<<grid(NROWS * NHID), 256, 0, stream>>>(scal, x_in, embW, embb, h, xws);

  const int edgeBlocks = BE / 64;     // 8128, exact
  const int nodeBlocks = NROWS / 64;  // 64, exact
  for (int l = 0; l < NLAYERS; ++l) {
    const float* b = wpk + l * LPK;
    const float* e1p = b;
    const float* e2p = b + SZ_E1;
    const float* x1p = b + SZ_E1 + SZ_SQ;
    const float* h1p = b + SZ_E1 + 2 * SZ_SQ;
    const float* h2p = b + SZ_E1 + 2 * SZ_SQ + SZ_H1;
    const int last = (l == NLAYERS - 1);

    zero_kernel<<<1, 256, 0, stream>>>(sums, 2 * NHP);
    zero_kernel<<<grid(NROWS * NHID), 256, 0, stream>>>(aggh, NROWS * NHID);
    if (!last) zero_kernel<<<grid(NROWS * 4), 256, 0, stream>>>(aggx, NROWS * 4);

    edge_pass1<<<edgeBlocks, 128, 0, stream>>>(h, xws, e1p, sums);
    bn_finalize<<<1, 128, 0, stream>>>(sums, L[l].ebng, L[l].ebnb, ac, 1.0f / (float)BE);
    edge_pass2<<<edgeBlocks, 128, 0, stream>>>(h, xws, e1p, ac, e2p, L[l].e2b,
                                               L[l].mW, L[l].mb, x1p, L[l].x1b, L[l].x2W,
                                               aggh, aggx, last);
    if (!last) x_update<<<grid(NROWS * 4), 256, 0, stream>>>(xws, aggx);

    zero_kernel<<<1, 256, 0, stream>>>(sums, 2 * NHP);
    node_pass1<<<nodeBlocks, 128, 0, stream>>>(h, aggh, scal, h1p, L[l].h1b, u, sums);
    bn_finalize<<<1, 128, 0, stream>>>(sums, L[l].hbng, L[l].hbnb, ac, 1.0f / (float)NROWS);
    node_pass2<<<nodeBlocks, 128, 0, stream>>>(u, ac, h2p, L[l].h2b, h);
  }

  decode_kernel<<<1, 256, 0, stream>>>(h, d1W, d1b, d2W, d2b, (float*)d_out);
  (void)out_size; (void)ws_size;
}